// ChamferDistance_32401233281613
// MI455X (gfx1250) — compile-verified
//
#include <hip/hip_runtime.h>
#include <hip/hip_bf16.h>

// ---------------------------------------------------------------------------
// ChamferDistance-style fused loss, MI455X (gfx1250, wave32).
// out = ( sum_i dist_i + sum_i |sqrt(tsdfOut_i) - tsdfGT_i| ) / (B*N)
//
// Bandwidth-bound: ~170 MB of read-once streams (point/tsdfOut/tsdfGT, loaded
// non-temporal so they don't thrash L2) + a gather into the 32^3 CP grid
// (393 KB/batch, 50 MB total -> L2-resident at 192 MB, regular-temporal).
// HBM floor ~170MB / 23.3 TB/s ~ 7.3 us.
//
// Index math is all uint32 (total = 8.4M << 2^31): 32-bit udiv lowers to a
// short rcp-based sequence instead of the 64-bit software-division loop.
// inUse index == i / NS exactly (N = NP*NS), so only two small udivs/element.
//
// Wave reductions use V_WMMA_F32_16X16X4_F32 (exact f32, deterministic).
// ---------------------------------------------------------------------------

typedef float v2f __attribute__((ext_vector_type(2)));
typedef float v8f __attribute__((ext_vector_type(8)));

// Sum of all 32 lanes' s via one f32 WMMA (B = ones => row sums of A).
// A(16x4): lanes 0-15 VGPR0 = A[m][0], lanes 16-31 VGPR0 = A[m][2]; VGPR1 = 0.
// => rowsum(m) = s[m] + s[m+16]. D(16x16) f32 layout: lane L<16 holds rows
// 0..7 of col L, lane L>=16 holds rows 8..15, so sum(c[0..7]) at lane 0 plus
// lane 16 is the full 32-lane sum.
__device__ __forceinline__ float wave_sum_wmma(float s) {
    v2f a; a[0] = s;    a[1] = 0.0f;
    v2f b; b[0] = 1.0f; b[1] = 1.0f;
    v8f c = {};
    c = __builtin_amdgcn_wmma_f32_16x16x4_f32(
            /*neg_a=*/false, a, /*neg_b=*/false, b,
            /*c_mod=*/(short)0, c, /*reuse_a=*/false, /*reuse_b=*/false);
    float t = c[0] + c[1] + c[2] + c[3] + c[4] + c[5] + c[6] + c[7];
    return __shfl(t, 0, 32) + __shfl(t, 16, 32);
}

// Block sum (<= 256 threads). Result valid on threadIdx.x == 0.
__device__ __forceinline__ float block_sum(float s) {
    __shared__ float smem[8];
    const int lane = threadIdx.x & 31;
    const int wid  = threadIdx.x >> 5;
    const float w = wave_sum_wmma(s);
    if (lane == 0) smem[wid] = w;
    __syncthreads();
    float r = 0.0f;
    if (threadIdx.x == 0) {
        const int nw = (blockDim.x + 31) >> 5;
        for (int i = 0; i < nw; ++i) r += smem[i];   // fixed order: deterministic
    }
    return r;
}

__global__ void cd_partial_kernel(const float* __restrict__ point,
                                  const float* __restrict__ CP,
                                  const float* __restrict__ tsdfOut,
                                  const float* __restrict__ tsdfGT,
                                  const int*   __restrict__ inUse,
                                  float* __restrict__ partials,
                                  unsigned N, unsigned NS, unsigned total) {
    const unsigned tid    = blockIdx.x * blockDim.x + threadIdx.x;
    const unsigned stride = gridDim.x * blockDim.x;

    float s = 0.0f;
    #pragma unroll 4
    for (unsigned i = tid; i < total; i += stride) {
        const unsigned b    = i / N;    // batch          (32-bit udiv: ~10 VALU)
        const unsigned prim = i / NS;   // == b*NP + n/NS (exact, since N = NP*NS)

        // point[i]: 3 consecutive floats; read-once stream -> non-temporal
        const float* p  = point + (size_t)i * 3;
        const float px = __builtin_nontemporal_load(p + 0);
        const float py = __builtin_nontemporal_load(p + 1);
        const float pz = __builtin_nontemporal_load(p + 2);

        // quantize: truncate toward zero (C cast == jnp astype(int32)), clamp [0,31]
        int ix = (int)((px + 0.5f) * 32.0f);
        int iy = (int)((py + 0.5f) * 32.0f);
        int iz = (int)((pz + 0.5f) * 32.0f);
        ix = min(max(ix, 0), 31);
        iy = min(max(iy, 0), 31);
        iz = min(max(iz, 0), 31);

        // gather from 32^3 grid: reused data, keep regular-temporal (L2-resident)
        const float* cp = CP + ((size_t)b * 32768u + (unsigned)(ix * 1024 + iy * 32 + iz)) * 3;
        const float dx = px - cp[0];
        const float dy = py - cp[1];
        const float dz = pz - cp[2];
        const float d  = sqrtf(dx * dx + dy * dy + dz * dz);

        // mask: when not in use, closest == point -> dist is exactly 0
        const float dist = (inUse[prim] == 1) ? d : 0.0f;

        // read-once streams -> non-temporal
        const float to = __builtin_nontemporal_load(tsdfOut + i);
        const float tg = __builtin_nontemporal_load(tsdfGT  + i);
        s += dist + fabsf(sqrtf(to) - tg);
    }

    // all lanes reconverged here: EXEC all-1s for the WMMA
    const float bs = block_sum(s);
    if (threadIdx.x == 0) partials[blockIdx.x] = bs;
}

__global__ void cd_final_kernel(const float* __restrict__ partials,
                                int nparts, float inv_count,
                                float* __restrict__ out) {
    float s = 0.0f;
    for (int i = threadIdx.x; i < nparts; i += blockDim.x) s += partials[i];
    const float t = block_sum(s);
    if (threadIdx.x == 0) out[0] = t * inv_count;
}

extern "C" void kernel_launch(void* const* d_in, const int* in_sizes, int n_in,
                              void* d_out, int out_size, void* d_ws, size_t ws_size,
                              hipStream_t stream) {
    const float* point   = (const float*)d_in[0];  // [B, NP, NS, 3]
    const float* CP      = (const float*)d_in[1];  // [B, 32, 32, 32, 3]
    const float* tsdfOut = (const float*)d_in[2];  // [B, NP*NS]
    const float* tsdfGT  = (const float*)d_in[3];  // [B, NP*NS]
    const int*   inUse   = (const int*)d_in[4];    // [B, NP]

    // derive shapes from flat sizes
    const int      B     = (int)((long long)in_sizes[1] / (32LL * 32 * 32 * 3));
    const unsigned N     = (unsigned)((long long)in_sizes[2] / B);  // NP*NS
    const int      NP    = in_sizes[4] / B;
    const unsigned NS    = N / (unsigned)NP;
    const unsigned total = (unsigned)B * N;       // 8,388,608 for the ref shapes

    const int NT = 256;                           // 8 wave32 per block
    int NB = 2048;                                // 524288 threads: 16 iters each, no tail
    if (ws_size < (size_t)NB * sizeof(float)) {
        NB = (int)(ws_size / sizeof(float));
        if (NB < 1) NB = 1;
    }
    float* partials = (float*)d_ws;

    cd_partial_kernel<<<NB, NT, 0, stream>>>(point, CP, tsdfOut, tsdfGT, inUse,
                                             partials, N, NS, total);

    const float inv_count = 1.0f / (float)total;
    cd_final_kernel<<<1, NT, 0, stream>>>(partials, NB, inv_count, (float*)d_out);
}